// SA_64381559767619
// MI455X (gfx1250) — compile-verified
//
#include <hip/hip_runtime.h>
#include <cmath>

typedef float v2f __attribute__((ext_vector_type(2)));
typedef float v8f __attribute__((ext_vector_type(8)));

#define EPSF 1e-5f

// Geometry
constexpr int BATCH  = 16;
constexpr int NPOINT = 4096;
constexpr int NSAMP  = 32;           // S
constexpr int C1 = 64, C2 = 64, C3 = 128;
constexpr int NPTS_TOTAL = BATCH * NPOINT;      // 65536 points
constexpr int WAVES_PER_WG = 8;                 // 256 threads
constexpr int HS = 66;                          // padded LDS row stride (floats)

// LDS layout (float offsets)
constexpr int W1_OFF  = 0;                       // 64  x HS
constexpr int W2_OFF  = W1_OFF + C2 * HS;        // 128 x HS
constexpr int W0_OFF  = W2_OFF + C3 * HS;        // 64 x 4 (K padded to 4)
constexpr int SC0_OFF = W0_OFF + C1 * 4;
constexpr int BI0_OFF = SC0_OFF + C1;
constexpr int SC1_OFF = BI0_OFF + C1;
constexpr int BI1_OFF = SC1_OFF + C2;
constexpr int SC2_OFF = BI1_OFF + C2;
constexpr int BI2_OFF = SC2_OFF + C3;
constexpr int H_OFF   = BI2_OFF + C3;            // 8 waves x (32 x HS)
constexpr int LDS_FLOATS = H_OFF + WAVES_PER_WG * NSAMP * HS;
constexpr size_t LDS_BYTES = (size_t)LDS_FLOATS * sizeof(float);

static __device__ __forceinline__ v8f wmma_f32(v2f a, v2f b, v8f c) {
    // exact fp32 matrix op: D(16x16) = A(16x4) * B(4x16) + C
    return __builtin_amdgcn_wmma_f32_16x16x4_f32(
        /*neg_a=*/false, a, /*neg_b=*/false, b,
        /*c_mod=*/(short)0, c, /*reuse_a=*/false, /*reuse_b=*/false);
}

__global__ void __launch_bounds__(256)
pointnet_sa_wmma_kernel(
    const float* __restrict__ xg,                       // [B,P,S,3]
    const float* __restrict__ w0, const float* __restrict__ b0,
    const float* __restrict__ g0, const float* __restrict__ be0,
    const float* __restrict__ mu0, const float* __restrict__ va0,
    const float* __restrict__ w1, const float* __restrict__ b1,
    const float* __restrict__ g1, const float* __restrict__ be1,
    const float* __restrict__ mu1, const float* __restrict__ va1,
    const float* __restrict__ w2, const float* __restrict__ b2,
    const float* __restrict__ g2, const float* __restrict__ be2,
    const float* __restrict__ mu2, const float* __restrict__ va2,
    float* __restrict__ feats)                          // [B,P,128]
{
    extern __shared__ float lds[];
    const int tid = threadIdx.x;

    // ---- Prologue: stage weights (stride-66 padded) + folded BN params ----
    for (int i = tid; i < C2 * C1; i += 256)            // w1 [o=64][c=64]
        lds[W1_OFF + (i >> 6) * HS + (i & 63)] = w1[i];
    for (int i = tid; i < C3 * C2; i += 256)            // w2 [o=128][c=64]
        lds[W2_OFF + (i >> 6) * HS + (i & 63)] = w2[i];
    if (tid < C1) {                                     // w0 [o=64][c=3] -> pad K=4
        lds[W0_OFF + tid * 4 + 0] = w0[tid * 3 + 0];
        lds[W0_OFF + tid * 4 + 1] = w0[tid * 3 + 1];
        lds[W0_OFF + tid * 4 + 2] = w0[tid * 3 + 2];
        lds[W0_OFF + tid * 4 + 3] = 0.0f;
        float s0 = g0[tid] * rsqrtf(va0[tid] + EPSF);
        lds[SC0_OFF + tid] = s0;
        lds[BI0_OFF + tid] = (b0[tid] - mu0[tid]) * s0 + be0[tid];
        float s1 = g1[tid] * rsqrtf(va1[tid] + EPSF);
        lds[SC1_OFF + tid] = s1;
        lds[BI1_OFF + tid] = (b1[tid] - mu1[tid]) * s1 + be1[tid];
    }
    if (tid < C3) {
        float s2 = g2[tid] * rsqrtf(va2[tid] + EPSF);
        lds[SC2_OFF + tid] = s2;
        lds[BI2_OFF + tid] = (b2[tid] - mu2[tid]) * s2 + be2[tid];
    }
    __syncthreads();

    // ---- One wave == one point (32 samples = two 16-row WMMA tiles) ----
    const int wave  = tid >> 5;
    const int lane  = tid & 31;
    const int halfv = lane >> 4;   // 0: K={0,1}, 1: K={2,3} for A/B frags
    const int l15   = lane & 15;
    const int point = blockIdx.x * WAVES_PER_WG + wave;

    float* h = lds + H_OFF + wave * (NSAMP * HS);
    const float* xp = xg + (size_t)point * (NSAMP * 3);

    // ---------------- Layer 0: [32x3(pad4)] x [4x16]x4 -> h [32x64] --------
    v2f a0[2];
#pragma unroll
    for (int m = 0; m < 2; ++m) {
        const int row = m * 16 + l15;
        const float* xr = xp + row * 3;
        a0[m].x = xr[halfv ? 2 : 0];
        a0[m].y = halfv ? 0.0f : xr[1];
    }
#pragma unroll
    for (int n = 0; n < 4; ++n) {
        const int col = n * 16 + l15;
        const v2f bfr = *(const v2f*)(&lds[W0_OFF + col * 4 + halfv * 2]);
        const float s = lds[SC0_OFF + col];
        const float t = lds[BI0_OFF + col];
#pragma unroll
        for (int m = 0; m < 2; ++m) {
            v8f acc = {0.f, 0.f, 0.f, 0.f, 0.f, 0.f, 0.f, 0.f};
            acc = wmma_f32(a0[m], bfr, acc);
            const int rowbase = m * 16 + halfv * 8;
#pragma unroll
            for (int r = 0; r < 8; ++r)
                h[(rowbase + r) * HS + col] = fmaxf(acc[r] * s + t, 0.0f);
        }
    }

    // ---------------- Layer 1: h[32x64] x w1 -> h[32x64] (in place) --------
#pragma unroll
    for (int m = 0; m < 2; ++m) {
        v2f af[16];
        {
            const int row = m * 16 + l15;
            const float* hr = h + row * HS + halfv * 2;
#pragma unroll
            for (int kk = 0; kk < 16; ++kk)
                af[kk] = *(const v2f*)(hr + kk * 4);    // ds_load_b64, aligned
        }
#pragma unroll
        for (int n = 0; n < 4; ++n) {
            const int col = n * 16 + l15;
            const float* wc = &lds[W1_OFF + col * HS + halfv * 2];
            v8f acc = {0.f, 0.f, 0.f, 0.f, 0.f, 0.f, 0.f, 0.f};
#pragma unroll
            for (int kk = 0; kk < 16; ++kk) {
                const v2f bfr = *(const v2f*)(wc + kk * 4);
                acc = wmma_f32(af[kk], bfr, acc);
            }
            const float s = lds[SC1_OFF + col];
            const float t = lds[BI1_OFF + col];
            const int rowbase = m * 16 + halfv * 8;
#pragma unroll
            for (int r = 0; r < 8; ++r)
                h[(rowbase + r) * HS + col] = fmaxf(acc[r] * s + t, 0.0f);
        }
    }

    // -------- Layer 2: h[32x64] x w2 -> [32x128], fused max over rows ------
    float nmax[8];
#pragma unroll
    for (int m = 0; m < 2; ++m) {
        v2f af[16];
        {
            const int row = m * 16 + l15;
            const float* hr = h + row * HS + halfv * 2;
#pragma unroll
            for (int kk = 0; kk < 16; ++kk)
                af[kk] = *(const v2f*)(hr + kk * 4);
        }
#pragma unroll
        for (int n = 0; n < 8; ++n) {
            const int col = n * 16 + l15;
            const float* wc = &lds[W2_OFF + col * HS + halfv * 2];
            v8f acc = {0.f, 0.f, 0.f, 0.f, 0.f, 0.f, 0.f, 0.f};
#pragma unroll
            for (int kk = 0; kk < 16; ++kk) {
                const v2f bfr = *(const v2f*)(wc + kk * 4);
                acc = wmma_f32(af[kk], bfr, acc);
            }
            const float s = lds[SC2_OFF + col];
            const float t = lds[BI2_OFF + col];
            float mx = 0.0f;                 // ReLU makes 0 the max identity
#pragma unroll
            for (int r = 0; r < 8; ++r)
                mx = fmaxf(mx, acc[r] * s + t);
            nmax[n] = (m == 0) ? mx : fmaxf(nmax[n], mx);
        }
    }

    // Combine lane halves (M rows split across lane^16) and store feats
#pragma unroll
    for (int n = 0; n < 8; ++n) {
        const float other = __shfl_xor(nmax[n], 16, 32);
        const float tot = fmaxf(nmax[n], other);
        if (lane < 16)
            feats[(size_t)point * C3 + n * 16 + l15] = tot;
    }
}

extern "C" void kernel_launch(void* const* d_in, const int* in_sizes, int n_in,
                              void* d_out, int out_size, void* d_ws, size_t ws_size,
                              hipStream_t stream) {
    (void)in_sizes; (void)n_in; (void)out_size; (void)d_ws; (void)ws_size;

    const float* xg   = (const float*)d_in[0];
    const float* newx = (const float*)d_in[1];
    const float* w0  = (const float*)d_in[2];
    const float* b0  = (const float*)d_in[3];
    const float* g0  = (const float*)d_in[4];
    const float* be0 = (const float*)d_in[5];
    const float* mu0 = (const float*)d_in[6];
    const float* va0 = (const float*)d_in[7];
    const float* w1  = (const float*)d_in[8];
    const float* b1  = (const float*)d_in[9];
    const float* g1  = (const float*)d_in[10];
    const float* be1 = (const float*)d_in[11];
    const float* mu1 = (const float*)d_in[12];
    const float* va1 = (const float*)d_in[13];
    const float* w2  = (const float*)d_in[14];
    const float* b2  = (const float*)d_in[15];
    const float* g2  = (const float*)d_in[16];
    const float* be2 = (const float*)d_in[17];
    const float* mu2 = (const float*)d_in[18];
    const float* va2 = (const float*)d_in[19];

    float* out = (float*)d_out;
    // Output 0: new_x passthrough [16,4096,3]
    hipMemcpyAsync(out, newx, (size_t)NPTS_TOTAL * 3 * sizeof(float),
                   hipMemcpyDeviceToDevice, stream);
    // Output 1: feats [16,4096,128]
    float* feats = out + (size_t)NPTS_TOTAL * 3;

    hipFuncSetAttribute((const void*)pointnet_sa_wmma_kernel,
                        hipFuncAttributeMaxDynamicSharedMemorySize,
                        (int)LDS_BYTES);

    const int grid = NPTS_TOTAL / WAVES_PER_WG;   // 8192 workgroups
    pointnet_sa_wmma_kernel<<<grid, WAVES_PER_WG * 32, LDS_BYTES, stream>>>(
        xg,
        w0, b0, g0, be0, mu0, va0,
        w1, b1, g1, be1, mu1, va1,
        w2, b2, g2, be2, mu2, va2,
        feats);
}